// SinkPITLossWrapper_50517405335905
// MI455X (gfx1250) — compile-verified
//
#include <hip/hip_runtime.h>
#include <hip/hip_bf16.h>
#include <math.h>

typedef __attribute__((ext_vector_type(2))) float v2f;
typedef __attribute__((ext_vector_type(8))) float v8f;

#define T_LEN   32000
#define B_SZ    64
#define N_SRC   8
#define EPS_F   1e-8f
#define BETA_F  10.0f
#define SINK_IT 100            // N_ITER/2 scan steps
#define CHUNK   1280           // K columns per block
#define NCHUNK  (T_LEN / CHUNK)   // 25
#define SPAN    (CHUNK / 8)       // 160 K per wave
// workspace layout per batch: 256 Gram + 16 rowsum = 272 floats
#define WSB     272

__global__ void zero_ws_kernel(float* ws, int n) {
    int i = blockIdx.x * blockDim.x + threadIdx.x;
    if (i < n) ws[i] = 0.0f;
}

// Batched Gram X*X^T via V_WMMA_F32_16X16X4_F32.
// X rows 0..7 = est[b], rows 8..15 = tgt[b]. Also accumulates row sums.
__global__ void gram_kernel(const float* __restrict__ est,
                            const float* __restrict__ tgt,
                            float* __restrict__ ws) {
    __shared__ float sG[WSB];   // [0,256) Gram partial, [256,272) row sums

    const int b     = blockIdx.x;
    const int chunk = blockIdx.y;
    const int tid   = threadIdx.x;
    const int lane  = tid & 31;
    const int wave  = tid >> 5;          // 0..7
    const int row   = lane & 15;         // M (and N) index of this lane
    const int khalf = (lane >> 4) * 2;   // 0 or 2: K sub-offset

    // zero LDS accumulators
    sG[tid] = 0.0f;
    if (tid < 16) sG[256 + tid] = 0.0f;
    __syncthreads();

    // row base pointer: rows 0-7 from est, 8-15 from tgt
    const float* rowptr = (row < 8)
        ? est + ((size_t)b * N_SRC + row) * T_LEN
        : tgt + ((size_t)b * N_SRC + (row - 8)) * T_LEN;

    const int k0 = chunk * CHUNK + wave * SPAN;

    v8f acc = {};
    float rsum = 0.0f;

#pragma unroll 4
    for (int k = 0; k < SPAN; k += 4) {
        // lane loads X[row][k0+k+khalf .. +1] : serves as both A and B operand
        v2f a = *(const v2f*)(rowptr + k0 + k + khalf);
        rsum += a.x + a.y;
        // D = A(16x4) * B(4x16) + C ; symmetric, so A-reg == B-reg
        acc = __builtin_amdgcn_wmma_f32_16x16x4_f32(
            /*neg_a=*/false, a, /*neg_b=*/false, a,
            /*c_mod=*/(short)0, acc, /*reuse_a=*/false, /*reuse_b=*/false);
    }

    // C layout: lane L holds N = L&15; VGPR r holds M = r + 8*(L>>4)
    const int N = row;
    const int Mbase = (lane >> 4) * 8;
#pragma unroll
    for (int r = 0; r < 8; ++r) {
        atomicAdd(&sG[(Mbase + r) * 16 + N], acc[r]);
    }
    atomicAdd(&sG[256 + row], rsum);
    __syncthreads();

    // flush block partial to global workspace
    float* gb = ws + (size_t)b * WSB;
    atomicAdd(&gb[tid], sG[tid]);
    if (tid < 16) atomicAdd(&gb[256 + tid], sG[256 + tid]);
}

// Per-batch: pairwise neg-SISDR from Gram stats, then Sinkhorn soft-perm.
// One block of 64 threads per batch; thread t owns matrix element (r,c).
__global__ void sinkhorn_kernel(const float* __restrict__ ws,
                                float* __restrict__ batch_loss) {
    __shared__ float pwS[64];
    __shared__ float Zs[64];
    __shared__ float red[64];

    const int b = blockIdx.x;
    const int t = threadIdx.x;
    const int r = t >> 3;     // first matrix index
    const int c = t & 7;      // second matrix index
    const float Tf = (float)T_LEN;

    const float* G = ws + (size_t)b * WSB;

    // pairwise neg-SISDR for (est=r, tgt=c)
    float me = G[256 + r] / Tf;
    float mt = G[256 + 8 + c] / Tf;
    float D  = G[r * 16 + 8 + c]        - Tf * me * mt;   // centered est_r . tgt_c
    float En = G[r * 16 + r]            - Tf * me * me;   // centered ||est_r||^2
    float Et = G[(8 + c) * 16 + 8 + c]  - Tf * mt * mt;   // centered ||tgt_c||^2
    float te = Et + EPS_F;
    float d2 = D * D;
    float p2 = d2 * Et / (te * te);                 // ||proj||^2
    float en2 = En - 2.0f * d2 / te + p2;           // ||e_noise||^2
    float sdr = p2 / (en2 + EPS_F);
    float pw  = -10.0f * log10f(sdr + EPS_F);       // pw_losses[b][r][c]

    pwS[r * 8 + c] = pw;
    __syncthreads();

    // C = transpose(pw): C[r][c] = pw[c][r]
    float Cv = pwS[c * 8 + r];
    float Z  = -BETA_F * Cv;

    for (int it = 0; it < SINK_IT; ++it) {
        // logsumexp over axis=1 (first index r), per column c
        Zs[r * 8 + c] = Z;
        __syncthreads();
        float m = -INFINITY;
#pragma unroll
        for (int k = 0; k < 8; ++k) m = fmaxf(m, Zs[k * 8 + c]);
        float s = 0.0f;
#pragma unroll
        for (int k = 0; k < 8; ++k) s += expf(Zs[k * 8 + c] - m);
        Z -= m + logf(s);
        __syncthreads();

        // logsumexp over axis=2 (second index c), per row r
        Zs[r * 8 + c] = Z;
        __syncthreads();
        float m2 = -INFINITY;
#pragma unroll
        for (int k = 0; k < 8; ++k) m2 = fmaxf(m2, Zs[r * 8 + k]);
        float s2 = 0.0f;
#pragma unroll
        for (int k = 0; k < 8; ++k) s2 += expf(Zs[r * 8 + k] - m2);
        Z -= m2 + logf(s2);
        __syncthreads();
    }

    // einsum('ij,ij->', C + Z/beta, exp(Z)) / n_src
    red[t] = (Cv + Z * (1.0f / BETA_F)) * expf(Z);
    __syncthreads();
#pragma unroll
    for (int stp = 32; stp > 0; stp >>= 1) {
        if (t < stp) red[t] += red[t + stp];
        __syncthreads();
    }
    if (t == 0) batch_loss[b] = red[0] * (1.0f / (float)N_SRC);
}

__global__ void mean_kernel(const float* __restrict__ bl, float* __restrict__ out) {
    __shared__ float red[B_SZ];
    int t = threadIdx.x;
    red[t] = bl[t];
    __syncthreads();
#pragma unroll
    for (int stp = 32; stp > 0; stp >>= 1) {
        if (t < stp) red[t] += red[t + stp];
        __syncthreads();
    }
    if (t == 0) out[0] = red[0] * (1.0f / (float)B_SZ);
}

extern "C" void kernel_launch(void* const* d_in, const int* in_sizes, int n_in,
                              void* d_out, int out_size, void* d_ws, size_t ws_size,
                              hipStream_t stream) {
    const float* est = (const float*)d_in[0];
    const float* tgt = (const float*)d_in[1];
    float* ws   = (float*)d_ws;
    float* gram = ws;                       // 64 * 272 floats
    float* bl   = ws + (size_t)B_SZ * WSB;  // 64 floats

    const int nz = B_SZ * WSB;
    zero_ws_kernel<<<(nz + 255) / 256, 256, 0, stream>>>(gram, nz);
    gram_kernel<<<dim3(B_SZ, NCHUNK), 256, 0, stream>>>(est, tgt, gram);
    sinkhorn_kernel<<<B_SZ, 64, 0, stream>>>(gram, bl);
    mean_kernel<<<1, B_SZ, 0, stream>>>(bl, (float*)d_out);
}